// LSTMModel_10823317586771
// MI455X (gfx1250) — compile-verified
//
#include <hip/hip_runtime.h>
#include <stdint.h>

typedef __attribute__((ext_vector_type(2))) float v2f;
typedef __attribute__((ext_vector_type(8))) float v8f;
typedef int v4i_gcc __attribute__((vector_size(16)));   // matches async builtin param type

#define B_ 256
#define T_ 1024
#define I_ 128
#define H_ 64
#define G_ 256   // 4*H
#define O_ 128
#define BT 16    // batch tile per workgroup
#define PAD 2    // stride ≡ 2 (mod 64): conflict-free strided b64 fragment reads

#if __has_builtin(__builtin_amdgcn_global_load_async_to_lds_b128) && \
    __has_builtin(__builtin_amdgcn_s_wait_asynccnt)
#define USE_ASYNC_X 1
#else
#define USE_ASYNC_X 0
#endif

typedef __attribute__((address_space(1))) v4i_gcc as1_v4i;
typedef __attribute__((address_space(3))) v4i_gcc as3_v4i;

__device__ __forceinline__ as1_v4i* to_global_v4(const void* p) {
  return (as1_v4i*)(uintptr_t)p;
}
__device__ __forceinline__ as3_v4i* to_lds_v4(void* p) {
  // generic pointers into LDS carry the LDS byte offset in their low 32 bits
  return (as3_v4i*)(uintptr_t)(uint32_t)(uintptr_t)p;
}

__device__ __forceinline__ float sigf(float x) {
  return 1.0f / (1.0f + __expf(-x));
}
__device__ __forceinline__ float tanh_fast(float x) {
  x = fminf(fmaxf(x, -15.0f), 15.0f);
  float e = __expf(-2.0f * x);
  return (1.0f - e) / (1.0f + e);
}

__global__ __launch_bounds__(256, 1)
void lstm2_fused_wmma(const float* __restrict__ x,
                      const float* __restrict__ h0,
                      const float* __restrict__ c0,
                      const float* __restrict__ w_ih0,
                      const float* __restrict__ w_hh0,
                      const float* __restrict__ b_ih0,
                      const float* __restrict__ b_hh0,
                      const float* __restrict__ w_ih1,
                      const float* __restrict__ w_hh1,
                      const float* __restrict__ b_ih1,
                      const float* __restrict__ b_hh1,
                      const float* __restrict__ w_fc,
                      const float* __restrict__ b_fc,
                      float* __restrict__ out) {
  // --- LDS: 319,360 B of the 327,680 B WGP budget ---
  __shared__ float sWhh0[G_][H_ + PAD];
  __shared__ float sWih1[G_][H_ + PAD];
  __shared__ float sWhh1[G_][H_ + PAD];
  __shared__ float sWih0h[128][I_ + PAD];   // w_ih0 rows 128..255 (u=1 gate tiles)
  __shared__ float sX[2][BT][I_ + PAD];     // double-buffered x tile
  __shared__ float sG[BT][G_ + PAD];
  __shared__ float sH0[BT][H_ + PAD];
  __shared__ float sC0[BT][H_ + PAD];
  __shared__ float sH1[BT][H_ + PAD];
  __shared__ float sC1[BT][H_ + PAD];

  const int tid  = threadIdx.x;
  const int wid  = tid >> 5;     // 8 waves
  const int lane = tid & 31;
  const int ln   = lane & 15;    // fragment row/col index
  const int kh   = lane >> 4;    // K-half select
  const int b0   = blockIdx.x * BT;

  // --- x streaming setup: this thread owns 8 floats (32 B) of the [BT, I] tile ---
  const int xr = (tid * 8) >> 7;        // row in tile
  const int xc = (tid * 8) & 127;       // col in tile
  const float* xbase = x + (size_t)(b0 + xr) * T_ * I_ + xc;

  // synchronous load of tile t=0 (latency overlapped with weight preload)
  float4 p0 = *(const float4*)(xbase);
  float4 p1 = *(const float4*)(xbase + 4);

  // --- preload resident weights into LDS (reused for all 1024 steps) ---
  for (int idx = tid; idx < G_ * H_; idx += 256) {
    int g = idx >> 6, k = idx & 63;
    sWhh0[g][k] = w_hh0[idx];
    sWih1[g][k] = w_ih1[idx];
    sWhh1[g][k] = w_hh1[idx];
  }
  for (int idx = tid; idx < 128 * I_; idx += 256) {
    int g = idx >> 7, k = idx & 127;
    sWih0h[g][k] = w_ih0[(size_t)(128 + g) * I_ + k];
  }
  // --- initial h/c state ---
  for (int e = tid; e < BT * H_; e += 256) {
    int r = e >> 6, j = e & 63;
    sH0[r][j] = h0[(b0 + r) * H_ + j];
    sC0[r][j] = c0[(b0 + r) * H_ + j];
    sH1[r][j] = h0[B_ * H_ + (b0 + r) * H_ + j];
    sC1[r][j] = c0[B_ * H_ + (b0 + r) * H_ + j];
  }

  // --- per-wave constants (hoisted out of the T loop) ---
  float bias0[2], bias1[2];
#pragma unroll
  for (int u = 0; u < 2; ++u) {
    int n0 = (wid + 8 * u) * 16;
    bias0[u] = b_ih0[n0 + ln] + b_hh0[n0 + ln];
    bias1[u] = b_ih1[n0 + ln] + b_hh1[n0 + ln];
  }
  // u=0 input-proj weights: global row pointer -> compiler promotes frags to VGPRs
  const float* wih0row0 = w_ih0 + (size_t)(wid * 16 + ln) * I_;
  // FC weights streamed from global (off the recurrence critical path; WGP$ hits)
  const float* wfcrow = w_fc + (size_t)(wid * 16 + ln) * H_;
  const float biasfc = b_fc[wid * 16 + ln];

  // stage tile t=0 into buffer 0
  *(float4*)&sX[0][xr][xc]     = p0;
  *(float4*)&sX[0][xr][xc + 4] = p1;
  __syncthreads();

#if !USE_ASYNC_X
  float4 nx0, nx1;
#endif

  int buf = 0;
  for (int t = 0; t < T_; ++t) {
    // --- stage x tile for t+1 into the alternate buffer.
    // Safe at iter top: all readers of sX[buf^1] passed the previous end barrier.
    {
      int tn = (t + 1 < T_) ? (t + 1) : (T_ - 1);
      const float* nsrc = xbase + (size_t)tn * I_;
#if USE_ASYNC_X
      // VGPR-free async copy, tracked by ASYNCcnt; completes during this iter.
      as1_v4i* gsrc = to_global_v4(nsrc);
      as3_v4i* ldst = to_lds_v4(&sX[buf ^ 1][xr][xc]);
      __builtin_amdgcn_global_load_async_to_lds_b128(gsrc, ldst, 0, 0);
      __builtin_amdgcn_global_load_async_to_lds_b128(gsrc, ldst, 16, 0);
#else
      nx0 = *(const float4*)(nsrc);
      nx1 = *(const float4*)(nsrc + 4);
#endif
      // pull the stream ~16 steps ahead of use into L2
      int tp = (t + 16 < T_) ? (t + 16) : (T_ - 1);
      __builtin_prefetch(xbase + (size_t)tp * I_, 0, 1);
    }

    // ---- layer 0 gates: xg = x_t @ w_ih0^T + (b_ih0+b_hh0) + h0 @ w_hh0^T ----
#pragma unroll
    for (int u = 0; u < 2; ++u) {
      const int n0 = (wid + 8 * u) * 16;
      const float bz = bias0[u];
      v8f acc = {bz, bz, bz, bz, bz, bz, bz, bz};
#pragma unroll
      for (int kk = 0; kk < I_ / 4; ++kk) {            // K = 128 (input proj)
        const int k = kk * 4 + 2 * kh;
        v2f a = *(const v2f*)&sX[buf][ln][k];
        v2f b = (u == 0) ? *(const v2f*)&wih0row0[k]              // VGPR-resident
                         : *(const v2f*)&sWih0h[n0 - 128 + ln][k]; // LDS-resident
        acc = __builtin_amdgcn_wmma_f32_16x16x4_f32(false, a, false, b,
                                                    (short)0, acc, false, false);
      }
#pragma unroll
      for (int kk = 0; kk < H_ / 4; ++kk) {            // K = 64 (recurrence)
        const int k = kk * 4 + 2 * kh;
        v2f a = *(const v2f*)&sH0[ln][k];
        v2f b = *(const v2f*)&sWhh0[n0 + ln][k];
        acc = __builtin_amdgcn_wmma_f32_16x16x4_f32(false, a, false, b,
                                                    (short)0, acc, false, false);
      }
#pragma unroll
      for (int v = 0; v < 8; ++v) sG[v + 8 * kh][n0 + ln] = acc[v];
    }
    __syncthreads();

    // ---- layer 0 elementwise ----
    for (int e = tid; e < BT * H_; e += 256) {
      int r = e >> 6, j = e & 63;
      float ig = sigf(sG[r][j]);
      float fg = sigf(sG[r][H_ + j]);
      float gg = tanh_fast(sG[r][2 * H_ + j]);
      float og = sigf(sG[r][3 * H_ + j]);
      float c  = fg * sC0[r][j] + ig * gg;
      sC0[r][j] = c;
      sH0[r][j] = og * tanh_fast(c);
    }
    __syncthreads();

    // ---- layer 1 gates: h0_t @ w_ih1^T + bias + h1 @ w_hh1^T ----
#pragma unroll
    for (int u = 0; u < 2; ++u) {
      const int n0 = (wid + 8 * u) * 16;
      const float bz = bias1[u];
      v8f acc = {bz, bz, bz, bz, bz, bz, bz, bz};
#pragma unroll
      for (int kk = 0; kk < H_ / 4; ++kk) {            // input proj (K = 64)
        const int k = kk * 4 + 2 * kh;
        v2f a = *(const v2f*)&sH0[ln][k];
        v2f b = *(const v2f*)&sWih1[n0 + ln][k];
        acc = __builtin_amdgcn_wmma_f32_16x16x4_f32(false, a, false, b,
                                                    (short)0, acc, false, false);
      }
#pragma unroll
      for (int kk = 0; kk < H_ / 4; ++kk) {            // recurrence (K = 64)
        const int k = kk * 4 + 2 * kh;
        v2f a = *(const v2f*)&sH1[ln][k];
        v2f b = *(const v2f*)&sWhh1[n0 + ln][k];
        acc = __builtin_amdgcn_wmma_f32_16x16x4_f32(false, a, false, b,
                                                    (short)0, acc, false, false);
      }
#pragma unroll
      for (int v = 0; v < 8; ++v) sG[v + 8 * kh][n0 + ln] = acc[v];
    }
    __syncthreads();

    // ---- layer 1 elementwise ----
    for (int e = tid; e < BT * H_; e += 256) {
      int r = e >> 6, j = e & 63;
      float ig = sigf(sG[r][j]);
      float fg = sigf(sG[r][H_ + j]);
      float gg = tanh_fast(sG[r][2 * H_ + j]);
      float og = sigf(sG[r][3 * H_ + j]);
      float c  = fg * sC1[r][j] + ig * gg;
      sC1[r][j] = c;
      sH1[r][j] = og * tanh_fast(c);
    }
    __syncthreads();

    // ---- fused FC for this timestep: logits[b, :] = h1 @ w_fc^T + b_fc ----
    {
      v8f acc = {biasfc, biasfc, biasfc, biasfc, biasfc, biasfc, biasfc, biasfc};
#pragma unroll
      for (int kk = 0; kk < H_ / 4; ++kk) {
        const int k = kk * 4 + 2 * kh;
        v2f a = *(const v2f*)&sH1[ln][k];
        v2f b = *(const v2f*)&wfcrow[k];
        acc = __builtin_amdgcn_wmma_f32_16x16x4_f32(false, a, false, b,
                                                    (short)0, acc, false, false);
      }
      const int n0 = wid * 16;   // 8 waves x 16 cols = O_ = 128
#pragma unroll
      for (int v = 0; v < 8; ++v) {
        int m = v + 8 * kh;
        out[((size_t)(b0 + m) * T_ + t) * O_ + n0 + ln] = acc[v];
      }
    }

    // ---- make the t+1 tile visible before the next iteration ----
#if USE_ASYNC_X
    __builtin_amdgcn_s_wait_asynccnt(0);   // copy was issued a full iter ago
#else
    *(float4*)&sX[buf ^ 1][xr][xc]     = nx0;
    *(float4*)&sX[buf ^ 1][xr][xc + 4] = nx1;
#endif
    __syncthreads();
    buf ^= 1;
  }

  // ---- final states: d_out = [logits | h_n(2,B,H) | c_n(2,B,H)] ----
  const size_t LOG = (size_t)B_ * T_ * O_;
  for (int e = tid; e < BT * H_; e += 256) {
    int r = e >> 6, j = e & 63;
    size_t off = (size_t)(b0 + r) * H_ + j;
    out[LOG + off]                       = sH0[r][j];
    out[LOG + (size_t)B_ * H_ + off]     = sH1[r][j];
    out[LOG + 2 * (size_t)B_ * H_ + off] = sC0[r][j];
    out[LOG + 3 * (size_t)B_ * H_ + off] = sC1[r][j];
  }
}

extern "C" void kernel_launch(void* const* d_in, const int* in_sizes, int n_in,
                              void* d_out, int out_size, void* d_ws, size_t ws_size,
                              hipStream_t stream) {
  (void)in_sizes; (void)n_in; (void)out_size; (void)d_ws; (void)ws_size;
  const float* x     = (const float*)d_in[0];
  const float* h0    = (const float*)d_in[1];
  const float* c0    = (const float*)d_in[2];
  const float* w_ih0 = (const float*)d_in[3];
  const float* w_hh0 = (const float*)d_in[4];
  const float* b_ih0 = (const float*)d_in[5];
  const float* b_hh0 = (const float*)d_in[6];
  const float* w_ih1 = (const float*)d_in[7];
  const float* w_hh1 = (const float*)d_in[8];
  const float* b_ih1 = (const float*)d_in[9];
  const float* b_hh1 = (const float*)d_in[10];
  const float* w_fc  = (const float*)d_in[11];
  const float* b_fc  = (const float*)d_in[12];
  float* out = (float*)d_out;

  lstm2_fused_wmma<<<dim3(B_ / BT), dim3(256), 0, stream>>>(
      x, h0, c0, w_ih0, w_hh0, b_ih0, b_hh0,
      w_ih1, w_hh1, b_ih1, b_hh1, w_fc, b_fc, out);
}